// HINT_80788334838380
// MI455X (gfx1250) — compile-verified
//
#include <hip/hip_runtime.h>
#include <hip/hip_bf16.h>

#define B_  8
#define L_  32
#define D_  16
#define DK_ 16
#define V_  874609
#define NT_ 4   // N-tiles (of 16 cols) per block in the logits GEMM

typedef __attribute__((ext_vector_type(2))) float v2f;
typedef __attribute__((ext_vector_type(8))) float v8f;

// ---------- half-wave (16-lane, == one row) reductions; wave32 ----------
__device__ __forceinline__ float hsum16(float v) {
    v += __shfl_xor(v, 8, 16);
    v += __shfl_xor(v, 4, 16);
    v += __shfl_xor(v, 2, 16);
    v += __shfl_xor(v, 1, 16);
    return v;
}
__device__ __forceinline__ float hmax16(float v) {
    v = fmaxf(v, __shfl_xor(v, 8, 16));
    v = fmaxf(v, __shfl_xor(v, 4, 16));
    v = fmaxf(v, __shfl_xor(v, 2, 16));
    v = fmaxf(v, __shfl_xor(v, 1, 16));
    return v;
}
// LayerNorm of one element of a 16-wide row held across 16 lanes
__device__ __forceinline__ float ln16(float x, float g, float b, float eps) {
    float mu  = hsum16(x) * 0.0625f;
    float xc  = x - mu;
    float var = hsum16(xc * xc) * 0.0625f;
    return xc * rsqrtf(var + eps) * g + b;
}

// ---------- Single-head attention layer on LDS-resident [32,16] ----------
// x: in/out hidden; nrm: scratch; q,k,v: projection buffers; att: [32,32]
// thread mapping: tid = row*16 + d
__device__ void sha_layer(float* x, float* nrm, float* q, float* k, float* v,
                          float* att,
                          const float* __restrict__ wq, const float* __restrict__ wk,
                          const float* __restrict__ wv, const float* __restrict__ wfc,
                          const float* __restrict__ lg, const float* __restrict__ lb,
                          int row, int d)
{
    const int rd = row * D_ + d;
    float xv = x[rd];                                // residual
    nrm[rd] = ln16(xv, lg[d], lb[d], 1e-6f);         // LN only on q path
    __syncthreads();

    float qa = 0.f, ka = 0.f, va = 0.f;
#pragma unroll
    for (int kk = 0; kk < D_; ++kk) {
        float xn = nrm[row * D_ + kk];
        float xr = x[row * D_ + kk];
        qa += xn * wq[kk * DK_ + d];
        ka += xr * wk[kk * DK_ + d];
        va += xr * wv[kk * DK_ + d];
    }
    q[rd] = qa; k[rd] = ka; v[rd] = va;
    __syncthreads();

    // scores for columns j = d and j = d+16, scaled by 1/sqrt(16)
    float s0 = 0.f, s1 = 0.f;
#pragma unroll
    for (int kk = 0; kk < DK_; ++kk) {
        float qv = q[row * DK_ + kk];
        s0 += qv * k[d * DK_ + kk];
        s1 += qv * k[(d + 16) * DK_ + kk];
    }
    s0 *= 0.25f; s1 *= 0.25f;
    float mx  = hmax16(fmaxf(s0, s1));
    float e0  = expf(s0 - mx), e1 = expf(s1 - mx);
    float inv = 1.0f / hsum16(e0 + e1);
    att[row * L_ + d]      = e0 * inv;
    att[row * L_ + d + 16] = e1 * inv;
    __syncthreads();

    float o = 0.f;
#pragma unroll
    for (int j = 0; j < L_; ++j) o += att[row * L_ + j] * v[j * DK_ + d];
    __syncthreads();
    nrm[rd] = o;                                     // reuse nrm as o-row buffer
    __syncthreads();

    float outv = xv;                                 // residual add
#pragma unroll
    for (int kk = 0; kk < DK_; ++kk) outv += nrm[row * DK_ + kk] * wfc[kk * D_ + d];
    __syncthreads();
    x[rd] = outv;
    __syncthreads();
}

// ---------- Kernel 1: everything except the big projection ----------
__global__ __launch_bounds__(512) void fused_preamble(
    const float* __restrict__ emb, const int* __restrict__ mask,
    const float* __restrict__ pad_vec,
    const float* __restrict__ wq, const float* __restrict__ wk,
    const float* __restrict__ wv, const float* __restrict__ wfc,
    const float* __restrict__ ln_g, const float* __restrict__ ln_b,
    const float* __restrict__ n1_g, const float* __restrict__ n1_b,
    const float* __restrict__ lin_g, const float* __restrict__ lin_b2,
    const float* __restrict__ nc_w, const float* __restrict__ nc_b,
    float* __restrict__ a_out, float* __restrict__ nl_out,
    float* __restrict__ lo_ws)
{
    __shared__ float sx[L_ * D_], snrm[L_ * D_], sq[L_ * D_], sk[L_ * D_],
                     sv[L_ * D_], skk[L_ * D_], skv[L_ * D_], satt[L_ * L_];
    const int b   = blockIdx.x;
    const int tid = threadIdx.x;
    const int row = tid >> 4;
    const int d   = tid & 15;
    const int rd  = tid;

    // ---- sequence pad: compact masked rows, fill the rest with pad_vec ----
    int mrow = mask[b * L_ + row];
    int pos  = 0;
    for (int j = 0; j < row; ++j) pos += (mask[b * L_ + j] != 0);
    sx[rd] = pad_vec[d];
    __syncthreads();
    if (mrow) sx[pos * D_ + d] = emb[(b * L_ + row) * D_ + d];
    __syncthreads();

    // ---- encoder: SHA0 -> LN -> SHA1 (save k,v) ----
    sha_layer(sx, snrm, sq, sk, sv, satt,
              wq + 0, wk + 0, wv + 0, wfc + 0, ln_g + 0, ln_b + 0, row, d);
    { float t = ln16(sx[rd], n1_g[d], n1_b[d], 1e-5f); sx[rd] = t; __syncthreads(); }
    sha_layer(sx, snrm, sq, skk, skv, satt,
              wq + 256, wk + 256, wv + 256, wfc + 256, ln_g + 16, ln_b + 16, row, d);

    // ---- decoder: SHA2 -> LN -> SHA3 (only q_ needed) ----
    sha_layer(sx, snrm, sq, sk, sv, satt,
              wq + 512, wk + 512, wv + 512, wfc + 512, ln_g + 32, ln_b + 32, row, d);
    { float t = ln16(sx[rd], n1_g[16 + d], n1_b[16 + d], 1e-5f); sx[rd] = t; __syncthreads(); }
    {
        float xv = sx[rd];
        snrm[rd] = ln16(xv, ln_g[48 + d], ln_b[48 + d], 1e-6f);
        __syncthreads();
        float qa = 0.f;
#pragma unroll
        for (int kk = 0; kk < D_; ++kk) qa += snrm[row * D_ + kk] * wq[768 + kk * DK_ + d];
        sq[rd] = qa;
        __syncthreads();
    }

    // ---- cross attention (temperature 1) against encoder k,v ----
    {
        float s0 = 0.f, s1 = 0.f;
#pragma unroll
        for (int kk = 0; kk < DK_; ++kk) {
            float qv = sq[row * DK_ + kk];
            s0 += qv * skk[d * DK_ + kk];
            s1 += qv * skk[(d + 16) * DK_ + kk];
        }
        float mx  = hmax16(fmaxf(s0, s1));
        float e0  = expf(s0 - mx), e1 = expf(s1 - mx);
        float inv = 1.0f / hsum16(e0 + e1);
        float p0 = e0 * inv, p1 = e1 * inv;
        satt[row * L_ + d]      = p0;
        satt[row * L_ + d + 16] = p1;
        a_out[(b * L_ + row) * L_ + d]      = p0;   // output 2: attention map
        a_out[(b * L_ + row) * L_ + d + 16] = p1;
        __syncthreads();
        float o = 0.f;
#pragma unroll
        for (int j = 0; j < L_; ++j) o += satt[row * L_ + j] * skv[j * DK_ + d];
        float lo = ln16(o, lin_g[d], lin_b2[d], 1e-5f);
        lo = lo > 0.f ? lo : 0.01f * lo;            // leaky relu
        lo_ws[(b * L_ + row) * D_ + d] = lo;        // A-matrix for the big GEMM
    }

    // ---- node_logits = leaky_relu(emb) @ nc_w + nc_b ----
    if (d < 3) {
        float acc = nc_b[d];
#pragma unroll
        for (int kk = 0; kk < D_; ++kk) {
            float e = emb[(b * L_ + row) * D_ + kk];
            e = e > 0.f ? e : 0.01f * e;
            acc += e * nc_w[kk * 3 + d];
        }
        nl_out[(b * L_ + row) * 3 + d] = acc;
    }
}

// ---------- Kernel 2: logits[256, V] = lo[256,16] @ lin_w[16,V] + bias ----------
// 16 waves/block; wave w owns M-tile w (rows w*16..w*16+15); block owns 4x16 N-cols.
// A fragments are loop-invariant over N: preload once, then per N-tile do
// 4x v_wmma_f32_16x16x4_f32 chained over K=16 and stream the D tile out with
// non-temporal stores (output is write-once, 896 MB >> 192 MB L2).
__global__ __launch_bounds__(512) void logits_wmma(
    const float* __restrict__ lo, const float* __restrict__ w,
    const float* __restrict__ bias, float* __restrict__ out)
{
    const int lane  = threadIdx.x & 31;
    const int wave  = threadIdx.x >> 5;        // m-tile 0..15
    const int nl    = lane & 15;
    const int n0    = blockIdx.x * (16 * NT_);
    const int khalf = (lane >> 4) << 1;        // lanes 0-15 -> K{0,1}; 16-31 -> K{2,3}
    const int mA    = (wave << 4) + nl;        // A row for this lane
    const int mbase = (wave << 4) + ((lane >> 4) << 3);  // D rows r / r+8 per lane half

    // preload the 4 A fragments (K chunks of 4) — invariant across N tiles
    v2f afrag[4];
#pragma unroll
    for (int i = 0; i < 4; ++i) {
        afrag[i].x = lo[mA * DK_ + 4 * i + khalf];
        afrag[i].y = lo[mA * DK_ + 4 * i + khalf + 1];
    }

#pragma unroll
    for (int t = 0; t < NT_; ++t) {
        const int n   = n0 + 16 * t + nl;
        const int ncl = n < V_ ? n : (V_ - 1);   // clamp loads; stores predicated

        v8f c = {0.f, 0.f, 0.f, 0.f, 0.f, 0.f, 0.f, 0.f};
#pragma unroll
        for (int i = 0; i < 4; ++i) {
            v2f b;
            b.x = w[(size_t)(4 * i + khalf) * V_ + ncl];
            b.y = w[(size_t)(4 * i + khalf + 1) * V_ + ncl];
            c = __builtin_amdgcn_wmma_f32_16x16x4_f32(
                    /*neg_a=*/false, afrag[i], /*neg_b=*/false, b,
                    /*c_mod=*/(short)0, c, /*reuse_a=*/false, /*reuse_b=*/false);
        }
        float bv = bias[ncl];
        if (n < V_) {
#pragma unroll
            for (int r = 0; r < 8; ++r)
                __builtin_nontemporal_store(c[r] + bv,
                                            out + (size_t)(mbase + r) * V_ + n);
        }
    }
}

extern "C" void kernel_launch(void* const* d_in, const int* in_sizes, int n_in,
                              void* d_out, int out_size, void* d_ws, size_t ws_size,
                              hipStream_t stream) {
    const float* emb     = (const float*)d_in[0];
    // d_in[1] = batch (unused; rows are sorted L per example)
    const int*   mask    = (const int*)d_in[2];
    const float* pad_vec = (const float*)d_in[3];
    const float* wq      = (const float*)d_in[4];
    const float* wk      = (const float*)d_in[5];
    const float* wv      = (const float*)d_in[6];
    const float* wfc     = (const float*)d_in[7];
    const float* ln_g    = (const float*)d_in[8];
    const float* ln_b    = (const float*)d_in[9];
    const float* n1_g    = (const float*)d_in[10];
    const float* n1_b    = (const float*)d_in[11];
    const float* lin_g   = (const float*)d_in[12];
    const float* lin_b2  = (const float*)d_in[13];
    const float* lin_w   = (const float*)d_in[14];
    const float* lin_bias= (const float*)d_in[15];
    const float* nc_w    = (const float*)d_in[16];
    const float* nc_b    = (const float*)d_in[17];

    float* out      = (float*)d_out;
    float* a_out    = out + (size_t)B_ * L_ * V_;
    float* nl_out   = a_out + (size_t)B_ * L_ * L_;
    float* lo_ws    = (float*)d_ws;                 // [256,16] f32 = 16 KB

    hipLaunchKernelGGL(fused_preamble, dim3(B_), dim3(512), 0, stream,
                       emb, mask, pad_vec, wq, wk, wv, wfc, ln_g, ln_b,
                       n1_g, n1_b, lin_g, lin_b2, nc_w, nc_b,
                       a_out, nl_out, lo_ws);

    const int nblocks = (V_ + (16 * NT_) - 1) / (16 * NT_);   // 13666 blocks
    hipLaunchKernelGGL(logits_wmma, dim3(nblocks), dim3(512), 0, stream,
                       lo_ws, lin_w, lin_bias, out);
}